// AdaFactorBackprop_33844342292855
// MI455X (gfx1250) — compile-verified
//
#include <hip/hip_runtime.h>
#include <math.h>

#define D      6144
#define BETA1  0.9f
#define BETA2  0.999f
#define EPS    1e-30f
#define STEP   0.01f

typedef __attribute__((ext_vector_type(2))) float v2f;
typedef __attribute__((ext_vector_type(4))) float v4f;
typedef __attribute__((ext_vector_type(8))) float v8f;

// ws layout (float offsets)
#define WS_RS    0                   // row-sum partials  [D][8]
#define WS_CS    (D*8)               // col-sum partials  [D][8]
#define WS_ROWF  (2*D*8)             // row_fact [D]
#define WS_COLF  (2*D*8 + D)         // col_fact [D]
#define WS_PART  (2*D*8 + 2*D)       // pass-2 partial sums [1024]
#define WS_INV   (WS_PART + 1024)    // scalar: STEP / clip_denom

// ---------------------------------------------------------------------------
// Row sums of grad^2 via V_WMMA_F32_16X16X4_F32 with B = ones:
// D[m,n] = sum_k A[m,k].  One wave per (16-row strip, 768-col chunk).
// Two accumulators (even/odd K chunks) break the WMMA RAW chain; the final
// element-wise merge is a fixed-order add -> deterministic.
// grad is left regular-temporal: 151 MB < 192 MB L2 and it is re-read 3 more
// times by later kernels.
// ---------------------------------------------------------------------------
__global__ void k_row_sums(const float* __restrict__ g, float* __restrict__ ws) {
    const int wave = blockIdx.x;          // 0..3071
    const int s    = wave >> 3;           // 16-row strip
    const int q    = wave & 7;            // 768-col chunk
    const int lane = threadIdx.x;         // wave32
    const int row  = s * 16 + (lane & 15);
    const long base = (long)row * D + q * 768 + ((lane >> 4) << 1);

    v8f c0 = {}, c1 = {};
    const v2f ones = {1.0f, 1.0f};
    for (int k = 0; k < 192; k += 2) {
        v2f ga = *(const v2f*)(g + base + k * 4);
        v2f gb = *(const v2f*)(g + base + k * 4 + 4);
        v2f a0 = ga * ga;
        v2f a1 = gb * gb;
        c0 = __builtin_amdgcn_wmma_f32_16x16x4_f32(false, a0, false, ones,
                                                   (short)0, c0, false, false);
        c1 = __builtin_amdgcn_wmma_f32_16x16x4_f32(false, a1, false, ones,
                                                   (short)0, c1, false, false);
    }
    v8f c = c0 + c1;
    // D[m,n] identical for all n; lane0 holds M=0..7 (N=0), lane16 holds M=8..15.
    if (lane == 0) {
        for (int j = 0; j < 8; ++j) ws[WS_RS + (s * 16 + j) * 8 + q] = c[j];
    } else if (lane == 16) {
        for (int j = 0; j < 8; ++j) ws[WS_RS + (s * 16 + 8 + j) * 8 + q] = c[j];
    }
}

// ---------------------------------------------------------------------------
// Col sums of grad^2: A = ones so D[m,n] = sum_k B[k,n].  Since A is all-ones,
// any bijective lane<->K assignment of B sums identically; we rely only on
// N = lane&15.  One wave per (16-col strip, 768-row chunk).
// ---------------------------------------------------------------------------
__global__ void k_col_sums(const float* __restrict__ g, float* __restrict__ ws) {
    const int wave = blockIdx.x;
    const int s    = wave >> 3;           // 16-col strip
    const int q    = wave & 7;            // 768-row chunk
    const int lane = threadIdx.x;
    const int col  = s * 16 + (lane & 15);
    const int rb   = q * 768 + ((lane >> 4) << 1);

    v8f c0 = {}, c1 = {};
    const v2f ones = {1.0f, 1.0f};
    for (int k = 0; k < 192; k += 2) {
        const int r0 = rb + k * 4;
        const int r1 = r0 + 4;
        float a  = g[(long)r0 * D + col];
        float b  = g[(long)(r0 + 1) * D + col];
        float cc = g[(long)r1 * D + col];
        float d  = g[(long)(r1 + 1) * D + col];
        v2f b0 = { a * a, b * b };
        v2f b1 = { cc * cc, d * d };
        c0 = __builtin_amdgcn_wmma_f32_16x16x4_f32(false, ones, false, b0,
                                                   (short)0, c0, false, false);
        c1 = __builtin_amdgcn_wmma_f32_16x16x4_f32(false, ones, false, b1,
                                                   (short)0, c1, false, false);
    }
    v8f c = c0 + c1;
    if (lane < 16) ws[WS_CS + col * 8 + q] = c[0];   // D[0, lane] = colsum(col)
}

// ---------------------------------------------------------------------------
// Fold partials -> row/col states, means, factors.  Single 1024-thread block.
// ---------------------------------------------------------------------------
__global__ void k_factors(const float* __restrict__ rowW, const float* __restrict__ colW,
                          float* __restrict__ ws) {
    __shared__ float red[1024];
    __shared__ float invr_s, invc_s;
    const int t = threadIdx.x;
    float rst[6], cst[6];
    float rsum = 0.f, csum = 0.f;
    const float invD = 1.0f / (float)D;

    for (int i = 0; i < 6; ++i) {
        const int r = t + i * 1024;
        float s = 0.f;
        for (int q = 0; q < 8; ++q) s += ws[WS_RS + r * 8 + q];
        float st = BETA2 * rowW[r] + (1.0f - BETA2) * (s * invD);
        rst[i] = st; rsum += st;
        s = 0.f;
        for (int q = 0; q < 8; ++q) s += ws[WS_CS + r * 8 + q];
        st = BETA2 * colW[r] + (1.0f - BETA2) * (s * invD);
        cst[i] = st; csum += st;
    }
    red[t] = rsum; __syncthreads();
    for (int o = 512; o > 0; o >>= 1) { if (t < o) red[t] += red[t + o]; __syncthreads(); }
    if (t == 0) invr_s = 1.0f / (red[0] * invD + EPS);
    __syncthreads();
    red[t] = csum; __syncthreads();
    for (int o = 512; o > 0; o >>= 1) { if (t < o) red[t] += red[t + o]; __syncthreads(); }
    if (t == 0) invc_s = 1.0f / (red[0] * invD + EPS);
    __syncthreads();
    const float invr = invr_s, invc = invc_s;
    for (int i = 0; i < 6; ++i) {
        const int r = t + i * 1024;
        ws[WS_ROWF + r] = sqrtf(rst[i] * invr);
        ws[WS_COLF + r] = sqrtf(cst[i] * invc);
    }
}

// ---------------------------------------------------------------------------
// Pass 2: sum(update^2) -> per-block partials (deterministic tree reduce).
// grad: regular (L2 hit from pass 1); exp_avg: non-temporal so it does not
// evict the L2-resident grad (grad still has 2 further reads).
// ---------------------------------------------------------------------------
__global__ void k_sumsq(const float* __restrict__ g, const float* __restrict__ ea,
                        float* __restrict__ ws) {
    __shared__ float red[256];
    const float* rowf = ws + WS_ROWF;
    const float* colf = ws + WS_COLF;
    float acc = 0.f;
    const int total4 = D * (D / 4);
    for (int i = blockIdx.x * blockDim.x + threadIdx.x; i < total4;
         i += gridDim.x * blockDim.x) {
        const int r  = i / (D / 4);
        const int c4 = (i - r * (D / 4)) * 4;
        const long off = (long)r * D + c4;
        v4f gv = *(const v4f*)(g + off);
        v4f ev = __builtin_nontemporal_load((const v4f*)(ea + off));
        v4f cf = *(const v4f*)(colf + c4);
        const float rf = rowf[r];
        v4f u = (BETA1 * ev + (1.0f - BETA1) * gv) / (rf * cf + EPS);
        acc += u.x * u.x + u.y * u.y + u.z * u.z + u.w * u.w;
    }
    red[threadIdx.x] = acc; __syncthreads();
    for (int o = 128; o > 0; o >>= 1) {
        if (threadIdx.x < o) red[threadIdx.x] += red[threadIdx.x + o];
        __syncthreads();
    }
    if (threadIdx.x == 0) ws[WS_PART + blockIdx.x] = red[0];
}

// ---------------------------------------------------------------------------
// Reduce pass-2 partials -> STEP/clip scalar; full vector step for b.
// ---------------------------------------------------------------------------
__global__ void k_scalar_and_b(const float* __restrict__ b, const float* __restrict__ gb,
                               const float* __restrict__ eab, const float* __restrict__ sqb,
                               float* __restrict__ outb, float* __restrict__ ws) {
    __shared__ float red[1024];
    __shared__ float bstep;
    const int t = threadIdx.x;

    red[t] = ws[WS_PART + t]; __syncthreads();
    for (int o = 512; o > 0; o >>= 1) { if (t < o) red[t] += red[t + o]; __syncthreads(); }
    if (t == 0) {
        float mean = red[0] / ((float)D * (float)D);
        ws[WS_INV] = STEP / fmaxf(1.0f, sqrtf(mean + EPS));
    }
    __syncthreads();

    float u[6]; float acc = 0.f;
    for (int i = 0; i < 6; ++i) {
        const int r = t + i * 1024;
        float gr = gb[r];
        float e  = BETA1 * eab[r] + (1.0f - BETA1) * gr;
        float s  = BETA2 * sqb[r] + (1.0f - BETA2) * gr * gr;
        u[i] = e / (sqrtf(s) + EPS);
        acc += u[i] * u[i];
    }
    red[t] = acc; __syncthreads();
    for (int o = 512; o > 0; o >>= 1) { if (t < o) red[t] += red[t + o]; __syncthreads(); }
    if (t == 0) {
        float mean = red[0] / (float)D;
        bstep = STEP / fmaxf(1.0f, sqrtf(mean + EPS));
    }
    __syncthreads();
    const float bs = bstep;
    for (int i = 0; i < 6; ++i) {
        const int r = t + i * 1024;
        outb[r] = b[r] - bs * u[i];
    }
}

// ---------------------------------------------------------------------------
// Pass 3: new_W = W - (STEP/clip) * update   (v4f streaming)
// Single-use streams (W in, new_W out, exp_avg) are non-temporal so they do
// not evict the L2-resident grad.
// ---------------------------------------------------------------------------
__global__ void k_apply(const float* __restrict__ W, const float* __restrict__ g,
                        const float* __restrict__ ea, float* __restrict__ outW,
                        const float* __restrict__ ws) {
    const float inv = ws[WS_INV];
    const float* rowf = ws + WS_ROWF;
    const float* colf = ws + WS_COLF;
    const int total4 = D * (D / 4);
    for (int i = blockIdx.x * blockDim.x + threadIdx.x; i < total4;
         i += gridDim.x * blockDim.x) {
        const int r  = i / (D / 4);
        const int c4 = (i - r * (D / 4)) * 4;
        const long off = (long)r * D + c4;
        v4f wv = __builtin_nontemporal_load((const v4f*)(W + off));
        v4f gv = *(const v4f*)(g + off);
        v4f ev = __builtin_nontemporal_load((const v4f*)(ea + off));
        v4f cf = *(const v4f*)(colf + c4);
        const float rf = rowf[r];
        v4f u = (BETA1 * ev + (1.0f - BETA1) * gv) / (rf * cf + EPS);
        v4f o = wv - inv * u;
        __builtin_nontemporal_store(o, (v4f*)(outW + off));
    }
}

extern "C" void kernel_launch(void* const* d_in, const int* in_sizes, int n_in,
                              void* d_out, int out_size, void* d_ws, size_t ws_size,
                              hipStream_t stream) {
    const float* W    = (const float*)d_in[0];
    const float* gW   = (const float*)d_in[1];
    const float* eaW  = (const float*)d_in[2];
    const float* rowW = (const float*)d_in[3];
    const float* colW = (const float*)d_in[4];
    const float* b    = (const float*)d_in[5];
    const float* gb   = (const float*)d_in[6];
    const float* eab  = (const float*)d_in[7];
    const float* sqb  = (const float*)d_in[8];

    float* outW = (float*)d_out;
    float* outb = outW + (size_t)D * D;
    float* ws   = (float*)d_ws;

    k_row_sums    <<<3072, 32,   0, stream>>>(gW, ws);
    k_col_sums    <<<3072, 32,   0, stream>>>(gW, ws);
    k_factors     <<<1,    1024, 0, stream>>>(rowW, colW, ws);
    k_sumsq       <<<1024, 256,  0, stream>>>(gW, eaW, ws);
    k_scalar_and_b<<<1,    1024, 0, stream>>>(b, gb, eab, sqb, outb, ws);
    k_apply       <<<4096, 256,  0, stream>>>(W, gW, eaW, outW, ws);
}